// BallQLoss_58377195487673
// MI455X (gfx1250) — compile-verified
//
#include <hip/hip_runtime.h>
#include <math.h>

// Problem constants (from reference setup_inputs)
#define NB     2
#define NPTS   8192
#define CMASK  30
#define K_BALL 16
#define R2     0.04f          // RADIUS^2 = 0.2^2

typedef __attribute__((ext_vector_type(2))) float v2f;
typedef __attribute__((ext_vector_type(8))) float v8f;

// One wave (32 lanes) owns a 16-query tile. It scans candidate tiles of 16
// points in ascending index order. The 16x16 dot-product block comes from
// V_WMMA_F32_16X16X4_F32 (A = queries 16x4 {x,y,z,0}, B = candidates 4x16).
// C/D layout: VGPR v, lanes 0-15 -> row M=v, lanes 16-31 -> row M=v+8,
// column N = lane&15. Ordered first-K selection per row via wave32 ballot.
__global__ __launch_bounds__(256)
void BallQLoss_kernel(const float* __restrict__ pc,
                      const float* __restrict__ mask,
                      float* __restrict__ out) {
    const int lane = threadIdx.x & 31;
    const int wave = threadIdx.x >> 5;
    const int tile = blockIdx.x * 8 + wave;         // 0..1023
    const int tilesPerBatch = NPTS / 16;            // 512
    const int b  = tile / tilesPerBatch;
    const int q0 = (tile % tilesPerBatch) * 16;
    const int h  = lane >> 4;                       // half-wave id
    const int n  = lane & 15;                       // column / row-in-tile

    const float* __restrict__ pcb   = pc   + (size_t)b * NPTS * 3;
    const float* __restrict__ maskb = mask + (size_t)b * NPTS * CMASK;

    // ---- A matrix (16x4 f32). Lane layout per ISA:
    //   lanes 0-15 : VGPR0 = K0(x), VGPR1 = K1(y)
    //   lanes16-31 : VGPR0 = K2(z), VGPR1 = K3(0)
    const float* qa = pcb + (size_t)(q0 + n) * 3;
    v2f a;
    a.x = h ? qa[2] : qa[0];
    a.y = h ? 0.0f  : qa[1];

    // Per-lane |q|^2 for the 8 rows this half-wave owns (rows v + 8h)
    float qn2[8];
#pragma unroll
    for (int v = 0; v < 8; ++v) {
        const float* p = pcb + (size_t)(q0 + v + 8 * h) * 3;
        float x = p[0], y = p[1], z = p[2];
        qn2[v] = x * x + y * y + z * z;
    }

    int   cnt[8] = {0, 0, 0, 0, 0, 0, 0, 0};
    float acc = 0.0f;

    for (int j0 = 0; j0 < NPTS; j0 += 16) {
        // Candidate column n = j0 + n
        const float* pj = pcb + (size_t)(j0 + n) * 3;
        float jx = pj[0], jy = pj[1], jz = pj[2];
        float cn2 = jx * jx + jy * jy + jz * jz;

        // ---- B matrix (4x16 f32), rows striped across lanes per VGPR:
        //   VGPR0: lanes 0-15 = K0(x), lanes 16-31 = K2(z)
        //   VGPR1: lanes 0-15 = K1(y), lanes 16-31 = K3(0)
        v2f bm;
        bm.x = h ? jz   : jx;
        bm.y = h ? 0.0f : jy;

        v8f c = {};
        c = __builtin_amdgcn_wmma_f32_16x16x4_f32(
                /*neg_a=*/false, a, /*neg_b=*/false, bm,
                /*c_mod=*/(short)0, c, /*reuse_a=*/false, /*reuse_b=*/false);

        if (j0 + 16 < NPTS)  // pull next pc tile toward the WGP
            __builtin_prefetch(pcb + (size_t)(j0 + 16 + n) * 3, 0, 3);

        bool allfull = true;
#pragma unroll
        for (int v = 0; v < 8; ++v) {
            float d2 = qn2[v] + cn2 - 2.0f * c[v];
            bool within = d2 < R2;
            unsigned bal     = __builtin_amdgcn_ballot_w32(within);
            unsigned rowbits = (bal >> (h * 16)) & 0xFFFFu;   // this row's 16 cols
            int rank = cnt[v] + __popc(rowbits & ((1u << n) - 1u));
            if (within && rank < K_BALL) {
                // Accepted neighbor: accumulate L1(mask_i, mask_j), 30 ch.
                const float* mi = maskb + (size_t)(q0 + v + 8 * h) * CMASK;
                const float* mj = maskb + (size_t)(j0 + n) * CMASK;
                float s = 0.0f;
#pragma unroll
                for (int cc = 0; cc < CMASK; cc += 2) {
                    float2 u = *(const float2*)(mi + cc);
                    float2 w = *(const float2*)(mj + cc);
                    s += fabsf(u.x - w.x) + fabsf(u.y - w.y);
                }
                acc += s;
            }
            cnt[v] += __popc(rowbits);
            allfull = allfull && (cnt[v] >= K_BALL);
        }
        // Wave-uniform early exit once every row has its 16 neighbors.
        if (__builtin_amdgcn_ballot_w32(allfull) == 0xFFFFFFFFu) break;
    }

    // ---- reduction: wave shuffle, then LDS across the 8 waves ----
#pragma unroll
    for (int off = 16; off > 0; off >>= 1)
        acc += __shfl_down(acc, off, 32);

    __shared__ float wsum[8];
    if (lane == 0) wsum[wave] = acc;
    __syncthreads();
    if (threadIdx.x == 0) {
        float t = 0.0f;
#pragma unroll
        for (int w = 0; w < 8; ++w) t += wsum[w];
        atomicAdd(out, t * (1.0f / ((float)NB * (float)NPTS * (float)K_BALL)));
    }
}

extern "C" void kernel_launch(void* const* d_in, const int* in_sizes, int n_in,
                              void* d_out, int out_size, void* d_ws, size_t ws_size,
                              hipStream_t stream) {
    (void)in_sizes; (void)n_in; (void)d_ws; (void)ws_size;
    const float* pc   = (const float*)d_in[0];
    const float* mask = (const float*)d_in[1];
    float* out = (float*)d_out;

    // d_out is poisoned by the harness; zero it on-stream (capture-legal).
    hipMemsetAsync(out, 0, sizeof(float) * (size_t)out_size, stream);

    const int totalTiles = NB * (NPTS / 16);     // 1024 waves of work
    dim3 grid(totalTiles / 8);                   // 8 waves / 256-thread block
    BallQLoss_kernel<<<grid, 256, 0, stream>>>(pc, mask, out);
}